// OHEM_Loss_58119497449808
// MI455X (gfx1250) — compile-verified
//
#include <hip/hip_runtime.h>

// ---------------- constants ----------------
#define R_TOTAL   8388608
#define CAP       2048      // == K in the reference
#define HALF_B    200       // BATCH/2
#define TPB1      256
#define VEC       4
#define TILES     8
#define TPB2      1024

// ---------------- CDNA5 async global->LDS helpers ----------------
// global_load_async_to_lds_b128: per-lane 16B copy, tracked by ASYNCcnt.
// LDS dest offset = low 32 bits of the flat shared pointer (AS3 offset).
__device__ __forceinline__ void async_ld_b128(const void* gaddr, void* lds_ptr) {
    unsigned loff = (unsigned)(unsigned long long)lds_ptr;
    asm volatile("global_load_async_to_lds_b128 %0, %1, off"
                 :: "v"(loff), "v"(gaddr) : "memory");
}

template <int N>
__device__ __forceinline__ void wait_asynccnt() {
    asm volatile("s_wait_asynccnt %0" :: "n"(N) : "memory");
}

// ---------------- math helpers (match reference FP op order) ----------------
__device__ __forceinline__ float ce_at(const float2* __restrict__ clsp, int idx, int cls) {
    float2 cp = clsp[idx];
    float m  = fmaxf(cp.x, cp.y);
    float e  = expf(cp.x - m) + expf(cp.y - m);
    float lse = m + logf(e);
    return lse - (cls ? cp.y : cp.x);          // -log_softmax[cls]
}

__device__ __forceinline__ float sl1_at(const float2* __restrict__ locp,
                                        const float2* __restrict__ loct, int idx) {
    float2 lp = locp[idx], lt = loct[idx];
    float d0 = fabsf(lt.x - lp.x), d1 = fabsf(lt.y - lp.y);
    float s0 = (d0 < 0.1f) ? 0.5f * d0 * d0 * 10.0f : d0 - 0.05f;
    float s1 = (d1 < 0.1f) ? 0.5f * d1 * d1 * 10.0f : d1 - 0.05f;
    return s0 + s1;
}

// float -> monotonically ascending u32 key
__device__ __forceinline__ unsigned fkey(float f) {
    unsigned u = __float_as_uint(f);
    return u ^ ((u & 0x80000000u) ? 0xFFFFFFFFu : 0x80000000u);
}

__device__ __forceinline__ float block_reduce(float v, float* buf, int t) {
    buf[t] = v;
    __syncthreads();
    for (int s = TPB2 / 2; s > 0; s >>= 1) {
        if (t < s) buf[t] += buf[t + s];
        __syncthreads();
    }
    float r = buf[0];
    __syncthreads();
    return r;
}

// ---------------- kernel 0: init counters ----------------
__global__ void init_kernel(unsigned* cnts) {
    if (threadIdx.x < 2) cnts[threadIdx.x] = 0u;
}

// ---------------- kernel 1: stream cls_target, compact valid indices ----------------
// Double-buffered async global->LDS pipeline (b128 per lane, ASYNCcnt-paced).
__global__ __launch_bounds__(TPB1) void scan_ct_kernel(const int* __restrict__ ct,
                                                       unsigned* __restrict__ cnts,
                                                       int* __restrict__ posIdx,
                                                       int* __restrict__ negIdx) {
    __shared__ int buf[2][TPB1 * VEC];
    const int t    = threadIdx.x;
    const int base = blockIdx.x * (TPB1 * VEC * TILES);

    // prologue: stage tile 0
    async_ld_b128(&ct[base + t * VEC], &buf[0][t * VEC]);

    for (int tile = 0; tile < TILES; ++tile) {
        const int cur = tile & 1;
        if (tile + 1 < TILES) {
            async_ld_b128(&ct[base + (tile + 1) * (TPB1 * VEC) + t * VEC],
                          &buf[cur ^ 1][t * VEC]);
            wait_asynccnt<1>();    // current tile's copy has landed (in-order)
        } else {
            wait_asynccnt<0>();
        }
#pragma unroll
        for (int v = 0; v < VEC; ++v) {
            int c = buf[cur][t * VEC + v];
            if ((unsigned)c <= 1u) {   // ct==0 (neg) or ct==1 (pos)
                int gidx = base + tile * (TPB1 * VEC) + t * VEC + v;
                unsigned slot = atomicAdd(&cnts[c ? 0 : 1], 1u);
                if (slot < CAP) (c ? posIdx : negIdx)[slot] = gidx;
            }
        }
    }
}

// ---------------- kernel 2: everything else on <=2048 candidates, single WG ----------------
__global__ __launch_bounds__(TPB2) void finalize_kernel(const float2* __restrict__ clsp,
                                                        const float2* __restrict__ locp,
                                                        const float2* __restrict__ loct,
                                                        const float4* __restrict__ anc,
                                                        const unsigned* __restrict__ cnts,
                                                        const int* __restrict__ posIdx,
                                                        const int* __restrict__ negIdx,
                                                        float* __restrict__ out) {
    // sort keys alias box arrays (keys dead after index extraction)
    __shared__ union {
        unsigned long long comp[CAP];
        struct { float x1[CAP]; float y1[CAP]; } b;
    } sh;
    __shared__ int           sidx[CAP];
    __shared__ float         bx2[CAP], by2[CAP];
    __shared__ float         sce[CAP], ssl[CAP];
    __shared__ unsigned char keep[CAP];
    __shared__ float         redbuf[TPB2];
    __shared__ float         resb[8];

    const int t = threadIdx.x;

    for (int ph = 0; ph < 2; ++ph) {                 // 0 = positives, 1 = negatives
        const int* idxArr = ph ? negIdx : posIdx;
        unsigned cnt = cnts[ph];
        const int n_cand = (cnt < (unsigned)CAP) ? (int)cnt : CAP;
        const int cls = ph ? 0 : 1;

        // 1) build sort keys: descending loss, ascending original index (top_k tie rule)
        for (int p = t; p < CAP; p += TPB2) {
            unsigned long long c = 0xFFFFFFFFFFFFFFFFull;   // padding sorts last
            if (p < n_cand) {
                int idx   = idxArr[p];
                float ce  = ce_at(clsp, idx, cls);
                float lss = ph ? ce : ce + sl1_at(locp, loct, idx);
                c = ((unsigned long long)(~fkey(lss)) << 32) | (unsigned)idx;
            }
            sh.comp[p] = c;
        }

        // 2) bitonic sort (ascending composite) over fixed 2048
        for (unsigned k = 2; k <= (unsigned)CAP; k <<= 1)
            for (unsigned j = k >> 1; j > 0; j >>= 1) {
                __syncthreads();
                for (unsigned i = t; i < (unsigned)CAP; i += TPB2) {
                    unsigned ixj = i ^ j;
                    if (ixj > i) {
                        unsigned long long a = sh.comp[i], b = sh.comp[ixj];
                        bool up = ((i & k) == 0);
                        if ((a > b) == up) { sh.comp[i] = b; sh.comp[ixj] = a; }
                    }
                }
            }
        __syncthreads();

        // 3) extract sorted indices (then comp storage is reusable)
        for (int p = t; p < CAP; p += TPB2)
            sidx[p] = (p < n_cand) ? (int)(unsigned)(sh.comp[p] & 0xFFFFFFFFull) : -1;
        __syncthreads();

        // 4) gather boxes + losses in sorted order; full-mask sums (deterministic order)
        float pce = 0.f, psl = 0.f;
        for (int p = t; p < CAP; p += TPB2) {
            unsigned char kp = 0;
            if (p < n_cand) {
                int idx = sidx[p];
                float4 a = anc[idx];
                sh.b.x1[p] = a.x; sh.b.y1[p] = a.y; bx2[p] = a.z; by2[p] = a.w;
                float ce = ce_at(clsp, idx, cls);
                float sl = sl1_at(locp, loct, idx);
                sce[p] = ce; ssl[p] = sl;
                pce += ce; psl += sl;
                kp = 1;
            }
            keep[p] = kp;
        }
        float fullce = block_reduce(pce, redbuf, t);
        float fullsl = block_reduce(psl, redbuf, t);

        // 5) NMS: sequential over kept order, parallel inner sweep
        for (int i = 0; i < n_cand; ++i) {
            __syncthreads();
            if (!keep[i]) continue;                         // uniform branch
            float X1 = sh.b.x1[i], Y1 = sh.b.y1[i], X2 = bx2[i], Y2 = by2[i];
            float areai = (X2 - X1) * (Y2 - Y1);
            for (int p = i + 1 + t; p < n_cand; p += TPB2) {
                if (keep[p]) {
                    float xx1 = fmaxf(X1, sh.b.x1[p]);
                    float yy1 = fmaxf(Y1, sh.b.y1[p]);
                    float xx2 = fminf(X2, bx2[p]);
                    float yy2 = fminf(Y2, by2[p]);
                    float inter = fmaxf(xx2 - xx1, 0.f) * fmaxf(yy2 - yy1, 0.f);
                    float areap = (bx2[p] - sh.b.x1[p]) * (by2[p] - sh.b.y1[p]);
                    float iou = inter / ((areai + areap) - inter);
                    if (iou > 0.7f) keep[p] = 0;
                }
            }
        }
        __syncthreads();

        // 6) rank / select (first HALF_B kept) — serial, n<=2048
        if (t == 0) {
            int run = 0; float selce = 0.f, selsl = 0.f;
            for (int p = 0; p < n_cand; ++p) {
                if (keep[p]) {
                    if (run < HALF_B) { selce += sce[p]; selsl += ssl[p]; }
                    ++run;
                }
            }
            if (ph == 0) {
                resb[0] = fullce; resb[1] = fullsl;
                resb[2] = selce;  resb[3] = selsl;
                resb[4] = (float)run;                      // nk (positives)
            } else {
                resb[5] = fullce; resb[6] = selce;
            }
        }
        __syncthreads();
    }

    if (t == 0) {
        int nk = (int)resb[4];
        bool trunc = nk > HALF_B;
        float pos_cls = trunc ? resb[2] : resb[0];
        float pos_loc = trunc ? resb[3] : resb[1];
        int keep_num = nk < HALF_B ? nk : HALF_B;

        int n_neg = (int)cnts[1];                          // true count (counter counts all)
        bool trunc_n = n_neg > HALF_B;
        float neg_cls = trunc_n ? resb[6] : resb[5];
        int keep_num_neg = n_neg < HALF_B ? n_neg : HALF_B;

        out[1] = pos_loc / (float)keep_num;                              // loc_loss
        out[0] = (neg_cls + pos_cls) / (float)(keep_num + keep_num_neg); // cls_loss
    }
}

// ---------------- host launcher ----------------
extern "C" void kernel_launch(void* const* d_in, const int* in_sizes, int n_in,
                              void* d_out, int out_size, void* d_ws, size_t ws_size,
                              hipStream_t stream) {
    (void)in_sizes; (void)n_in; (void)out_size; (void)ws_size;

    const float2* cls_pred   = (const float2*)d_in[0];   // (1,R,2) f32
    const int*    cls_target = (const int*)d_in[1];      // (1,1,R) i32
    const float2* loc_pred   = (const float2*)d_in[2];   // (1,R,2) f32
    const float2* loc_target = (const float2*)d_in[3];   // (1,R,2) f32
    const float4* anchors    = (const float4*)d_in[4];   // (1,R,4) f32
    float*        out        = (float*)d_out;            // [cls_loss, loc_loss]

    unsigned* cnts   = (unsigned*)d_ws;                  // [0]=pos count, [1]=neg count
    int*      posIdx = (int*)d_ws + 16;                  // 64B-aligned region
    int*      negIdx = posIdx + CAP;

    init_kernel<<<1, 32, 0, stream>>>(cnts);

    const int blocks = R_TOTAL / (TPB1 * VEC * TILES);   // 1024
    scan_ct_kernel<<<blocks, TPB1, 0, stream>>>(cls_target, cnts, posIdx, negIdx);

    finalize_kernel<<<1, TPB2, 0, stream>>>(cls_pred, loc_pred, loc_target, anchors,
                                            cnts, posIdx, negIdx, out);
}